// PointPillarScatter_Seg_42107859370503
// MI455X (gfx1250) — compile-verified
//
#include <hip/hip_runtime.h>
#include <hip/hip_bf16.h>

// Problem constants from the reference (PointPillarScatter).
constexpr int C    = 64;
constexpr int NX   = 512;
constexpr int NY   = 512;
constexpr int NYNX = NY * NX;

constexpr int PPB   = 16;        // pillars per block
constexpr int TPP   = 16;        // threads per pillar (4 channels each)
constexpr int BLOCK = PPB * TPP; // 256 threads = 8 waves (wave32)

// ---------------- zero-fill: 256 MB of b128 stores ----------------
__global__ void ppscatter_fill_zero(float4* __restrict__ out, long long n4) {
    long long i      = (long long)blockIdx.x * blockDim.x + threadIdx.x;
    long long stride = (long long)gridDim.x * blockDim.x;
    float4 z = make_float4(0.f, 0.f, 0.f, 0.f);
    for (; i < n4; i += stride) out[i] = z;
}

// ---------------- gfx1250 async global->LDS helpers ----------------
typedef int v4i_b128 __attribute__((vector_size(16)));  // matches builtin param
typedef __attribute__((address_space(1))) v4i_b128* gptr_b128;
typedef __attribute__((address_space(3))) v4i_b128* lptr_b128;

__device__ inline void async_ld_b128(const void* gsrc, void* ldst) {
#if __has_builtin(__builtin_amdgcn_global_load_async_to_lds_b128)
    __builtin_amdgcn_global_load_async_to_lds_b128(
        (gptr_b128)gsrc, (lptr_b128)ldst, /*imm offset=*/0, /*cpol=*/0);
#else
    unsigned lds_off = (unsigned)(unsigned long long)
        (__attribute__((address_space(3))) char*)ldst;
    asm volatile("global_load_async_to_lds_b128 %0, %1, off"
                 :: "v"(lds_off), "v"(gsrc) : "memory");
#endif
}

__device__ inline void wait_async0() {
#if __has_builtin(__builtin_amdgcn_s_wait_asynccnt)
    __builtin_amdgcn_s_wait_asynccnt(0);
#else
    asm volatile("s_wait_asynccnt 0x0" ::: "memory");
#endif
}

// ---------------- scatter ----------------
// out[b, c, y, x] = feats[p, c]  where coords[p] = (b, z, y, x)
__global__ void ppscatter_scatter(const float* __restrict__ feats,
                                  const int* __restrict__ coords,
                                  float* __restrict__ out, int P) {
    __shared__ alignas(16) int s_coords[PPB * 4];

    const int p0  = blockIdx.x * PPB;
    const int tid = threadIdx.x;

    // Stage this block's 16 coord rows (16 B each) into LDS via the async
    // global->LDS path: one b128 async copy per pillar, issued by lanes 0..15.
    if (tid < PPB && (p0 + tid) < P) {
        async_ld_b128(coords + (size_t)(p0 + tid) * 4, &s_coords[tid * 4]);
    }

    const int lp = tid >> 4;   // local pillar 0..15
    const int t  = tid & 15;   // channel sub-group 0..15 (4 channels each)
    const int p  = p0 + lp;

    // Coalesced feature read (16 lanes x float4 = one 256B row), issued
    // before the async wait so it overlaps the coord staging.
    float4 f = make_float4(0.f, 0.f, 0.f, 0.f);
    if (p < P) {
        f = *(const float4*)(feats + (size_t)p * C + (size_t)t * 4);
    }

    wait_async0();      // ASYNCcnt==0 in the issuing wave; free elsewhere
    __syncthreads();    // make LDS coords visible to all 8 waves

    if (p < P) {
        const int b = s_coords[lp * 4 + 0];
        const int y = s_coords[lp * 4 + 2];
        const int x = s_coords[lp * 4 + 3];
        const size_t base = (size_t)b * ((size_t)C * NYNX)
                          + (size_t)y * NX + (size_t)x;
        const int c0 = t * 4;
        // Channel stride is NYNX floats (1 MB) -> 4 scalar scattered stores.
        out[base + (size_t)(c0 + 0) * NYNX] = f.x;
        out[base + (size_t)(c0 + 1) * NYNX] = f.y;
        out[base + (size_t)(c0 + 2) * NYNX] = f.z;
        out[base + (size_t)(c0 + 3) * NYNX] = f.w;
    }
}

extern "C" void kernel_launch(void* const* d_in, const int* in_sizes, int n_in,
                              void* d_out, int out_size, void* d_ws, size_t ws_size,
                              hipStream_t stream) {
    const float* feats  = (const float*)d_in[0];   // [P, 64] f32
    const int*   coords = (const int*)d_in[1];     // [P, 4]  i32 (b, z, y, x)
    float*       out    = (float*)d_out;           // [B, 64, 512, 512] f32

    const int P = in_sizes[1] / 4;

    // 1) zero the canvas (dominant traffic: 256 MB of b128 stores)
    const long long n4 = (long long)out_size / 4;
    ppscatter_fill_zero<<<8192, 256, 0, stream>>>((float4*)d_out, n4);

    // 2) scatter pillar features
    const int blocks = (P + PPB - 1) / PPB;
    ppscatter_scatter<<<blocks, BLOCK, 0, stream>>>(feats, coords, out, P);
}